// QuanvolutionNEQR_9526237463003
// MI455X (gfx1250) — compile-verified
//
#include <hip/hip_runtime.h>

typedef float v2f __attribute__((ext_vector_type(2)));
typedef float v8f __attribute__((ext_vector_type(8)));

#define NQ 10
#define DIM 1024
#define OW 27
#define OH 27
#define BATCH 16
#define NPATCH (BATCH * OW * OH)   // 11664
#define WPB 2                      // patches (waves) per workgroup

// workspace layout (float offsets)
#define WS_PC 0      // [2][1024] cos(phase)
#define WS_PS 2048   // [2][1024] sin(phase)
#define WS_A  4096   // [2][32x32] Kron(R(ry0..ry4))
#define WS_B  6144   // [2][32x32] Kron(R(ry5..ry9))

// ---------------------------------------------------------------------------
// Setup: fuse the CRZ ring into one diagonal phase per block, and collapse the
// 10 commuting RY layers into two 32x32 Kronecker factors A (qubits 0-4) and
// B (qubits 5-9).  One thread per (block, element) pair; 2*1024 threads.
// ---------------------------------------------------------------------------
__global__ void quanv_setup(const float* __restrict__ kern, float* __restrict__ ws) {
  int tid = blockIdx.x * blockDim.x + threadIdx.x;
  if (tid >= 2 * DIM) return;
  int blk = tid >> 10;
  int d   = tid & (DIM - 1);
  const float* crz = kern + blk * 2 * NQ;
  const float* ry  = crz + NQ;

  // phase_exp[d] = sum_i qb[d,i] * (2*qb[d,(i+1)%10] - 1) * crz[i]/2
  float pe = 0.f;
#pragma unroll
  for (int i = 0; i < NQ; ++i) {
    int it = (i + 1) % NQ;
    int qi = (d >> (NQ - 1 - i)) & 1;
    int qt = (d >> (NQ - 1 - it)) & 1;
    pe += (float)qi * (2.f * (float)qt - 1.f) * (crz[i] * 0.5f);
  }
  ws[WS_PC + blk * DIM + d] = cosf(pe);
  ws[WS_PS + blk * DIM + d] = sinf(pe);

  // A[rp][r] = prod_{b=0..4} R(ry[b])[bit(rp,4-b)][bit(r,4-b)]
  // B[rp][r] = prod_{b=0..4} R(ry[5+b])[bit(rp,4-b)][bit(r,4-b)]
  // R(t) = [[c,-s],[s,c]], c=cos(t/2), s=sin(t/2)
  int rp = d >> 5, r = d & 31;
  float a = 1.f, bm = 1.f;
#pragma unroll
  for (int b = 0; b < 5; ++b) {
    int ib = (rp >> (4 - b)) & 1;
    int jb = (r  >> (4 - b)) & 1;
    float h = ry[b] * 0.5f;
    float c = cosf(h), s = sinf(h);
    a *= ib ? (jb ? c : s) : (jb ? -s : c);
    float h2 = ry[5 + b] * 0.5f;
    float c2 = cosf(h2), s2 = sinf(h2);
    bm *= ib ? (jb ? c2 : s2) : (jb ? -s2 : c2);
  }
  ws[WS_A + blk * DIM + d] = a;
  ws[WS_B + blk * DIM + d] = bm;
}

// ---------------------------------------------------------------------------
// Main: one wave per patch.  State = 32x32 complex matrix in LDS (row = qubits
// 0-4, col = qubits 5-9).  Per block: diagonal phase, then S <- A*S*B^T done as
// two chained 32x32x32 f32 GEMMs per component using V_WMMA_F32_16X16X4_F32.
// Finally |psi|^2 reduced against per-qubit Z signs via wave shuffles.
// ---------------------------------------------------------------------------
__global__ void __launch_bounds__(WPB * 32) quanv_main(const int* __restrict__ img,
                                                       const float* __restrict__ ws,
                                                       float* __restrict__ out) {
  __shared__ float Ssh[WPB][2][32][33];  // [wave][re/im][row][col+pad]
  __shared__ float Tsh[WPB][2][32][33];

  const int lane = threadIdx.x & 31;
  const int wid  = threadIdx.x >> 5;
  const int p    = blockIdx.x * WPB + wid;
  if (p >= NPATCH) return;  // wave-uniform

  float (*Sre)[33] = Ssh[wid][0];
  float (*Sim)[33] = Ssh[wid][1];
  float (*Tre)[33] = Tsh[wid][0];
  float (*Tim)[33] = Tsh[wid][1];

  // zero state
#pragma unroll 4
  for (int t = 0; t < 32; ++t) { Sre[t][lane] = 0.f; Sim[t][lane] = 0.f; }

  // NEQR init: amplitude 0.5 at q*256 + bitrev8(pixel_q), q = di*2+dj
  {
    int b   = p / (OW * OH);
    int rem = p - b * OW * OH;
    int i   = rem / OH;
    int j   = rem - i * OH;
    if (lane < 4) {
      int di = lane >> 1, dj = lane & 1;
      unsigned v = (unsigned)img[(b * 28 + (i + di)) * 28 + (j + dj)] & 255u;
      v = ((v & 0xF0u) >> 4) | ((v & 0x0Fu) << 4);
      v = ((v & 0xCCu) >> 2) | ((v & 0x33u) << 2);
      v = ((v & 0xAAu) >> 1) | ((v & 0x55u) << 1);
      int idx = lane * 256 + (int)v;          // bits 9..5 = row, 4..0 = col
      Sre[idx >> 5][idx & 31] = 0.5f;
    }
  }

  const int M  = lane & 15;   // fragment row (A/D) or col (B/D)
  const int hi = lane >> 4;   // half-wave: covers K = 2..3 of each K4 step

#pragma unroll 1
  for (int blk = 0; blk < 2; ++blk) {
    const float* pc  = ws + WS_PC + blk * DIM;
    const float* psn = ws + WS_PS + blk * DIM;
    const float* Am  = ws + WS_A  + blk * DIM;
    const float* Bm  = ws + WS_B  + blk * DIM;

    // fused CRZ diagonal: psi *= exp(i*phase)
#pragma unroll 4
    for (int t = 0; t < 32; ++t) {
      int e = t * 32 + lane;
      float c = pc[e], s = psn[e];
      float re = Sre[t][lane], im = Sim[t][lane];
      Sre[t][lane] = re * c - im * s;
      Sim[t][lane] = re * s + im * c;
    }

    // stage 1: T = A * S   (per component)
#pragma unroll 1
    for (int part = 0; part < 2; ++part) {
      float (*Sp)[33] = part ? Sim : Sre;
      float (*Tp)[33] = part ? Tim : Tre;
#pragma unroll 1
      for (int ti = 0; ti < 2; ++ti) {
#pragma unroll 1
        for (int tj = 0; tj < 2; ++tj) {
          v8f acc = {};
#pragma unroll
          for (int k0 = 0; k0 < 32; k0 += 4) {
            int ka = k0 + hi * 2;
            v2f af, bf;
            af.x = Am[(ti * 16 + M) * 32 + ka];      // A[m][k],  A[m][k+1]
            af.y = Am[(ti * 16 + M) * 32 + ka + 1];
            bf.x = Sp[ka][tj * 16 + M];              // S[k][n],  S[k+1][n]
            bf.y = Sp[ka + 1][tj * 16 + M];
            acc = __builtin_amdgcn_wmma_f32_16x16x4_f32(
                false, af, false, bf, (short)0, acc, false, false);
          }
#pragma unroll
          for (int v = 0; v < 8; ++v)                // D: M = v (+8 for hi)
            Tp[ti * 16 + hi * 8 + v][tj * 16 + M] = acc[v];
        }
      }
    }

    // stage 2: S = T * B^T  (B-operand loaded index-swapped: B[n][k])
#pragma unroll 1
    for (int part = 0; part < 2; ++part) {
      float (*Sp)[33] = part ? Sim : Sre;
      float (*Tp)[33] = part ? Tim : Tre;
#pragma unroll 1
      for (int ti = 0; ti < 2; ++ti) {
#pragma unroll 1
        for (int tj = 0; tj < 2; ++tj) {
          v8f acc = {};
#pragma unroll
          for (int k0 = 0; k0 < 32; k0 += 4) {
            int ka = k0 + hi * 2;
            v2f af, bf;
            af.x = Tp[ti * 16 + M][ka];              // T[m][k]
            af.y = Tp[ti * 16 + M][ka + 1];
            bf.x = Bm[(tj * 16 + M) * 32 + ka];      // B^T[k][n] = B[n][k]
            bf.y = Bm[(tj * 16 + M) * 32 + ka + 1];
            acc = __builtin_amdgcn_wmma_f32_16x16x4_f32(
                false, af, false, bf, (short)0, acc, false, false);
          }
#pragma unroll
          for (int v = 0; v < 8; ++v)
            Sp[ti * 16 + hi * 8 + v][tj * 16 + M] = acc[v];
        }
      }
    }
  }

  // z[q] = sum_d (1 - 2*bit(d, 9-q)) * |psi_d|^2
  float z[NQ];
#pragma unroll
  for (int q = 0; q < NQ; ++q) z[q] = 0.f;
#pragma unroll 4
  for (int t = 0; t < 32; ++t) {
    int e = t * 32 + lane;
    float re = Sre[t][lane], im = Sim[t][lane];
    float pr = re * re + im * im;
#pragma unroll
    for (int q = 0; q < NQ; ++q)
      z[q] += ((e >> (NQ - 1 - q)) & 1) ? -pr : pr;
  }
#pragma unroll
  for (int off = 16; off > 0; off >>= 1) {
#pragma unroll
    for (int q = 0; q < NQ; ++q)
      z[q] += __shfl_xor(z[q], off, 32);
  }
  if (lane == 0) {
#pragma unroll
    for (int q = 0; q < NQ; ++q)
      out[p * NQ + q] = z[q];
  }
}

// ---------------------------------------------------------------------------
extern "C" void kernel_launch(void* const* d_in, const int* in_sizes, int n_in,
                              void* d_out, int out_size, void* d_ws, size_t ws_size,
                              hipStream_t stream) {
  (void)in_sizes; (void)n_in; (void)out_size; (void)ws_size;
  const int*   img  = (const int*)d_in[0];
  const float* kern = (const float*)d_in[1];
  float*       ws   = (float*)d_ws;    // needs 8192 floats = 32 KB
  float*       out  = (float*)d_out;

  quanv_setup<<<8, 256, 0, stream>>>(kern, ws);
  quanv_main<<<(NPATCH + WPB - 1) / WPB, WPB * 32, 0, stream>>>(img, ws, out);
}